// GCNV2_30021821399140
// MI455X (gfx1250) — compile-verified
//
#include <hip/hip_runtime.h>
#include <cstdint>

#define SLOPE 0.2f
#define BN_EPS 1e-5f

typedef float v2f __attribute__((ext_vector_type(2)));
typedef float v8f __attribute__((ext_vector_type(8)));

// ---------------- generic zero (float4) ----------------
__global__ void zero4_kernel(float4* __restrict__ p, long long n4) {
  long long i = (long long)blockIdx.x * blockDim.x + threadIdx.x;
  if (i < n4) p[i] = make_float4(0.f, 0.f, 0.f, 0.f);
}

// ---------------- degree / norms ----------------
__global__ void degree_kernel(const int* __restrict__ src, const int* __restrict__ dst,
                              float* __restrict__ deg_out, float* __restrict__ deg_in, int E) {
  int e = blockIdx.x * blockDim.x + threadIdx.x;
  if (e >= E) return;
  unsafeAtomicAdd(&deg_out[src[e]], 1.0f);
  unsafeAtomicAdd(&deg_in[dst[e]], 1.0f);
}

__global__ void norm_kernel(float* __restrict__ d, long long n) {
  long long i = (long long)blockIdx.x * blockDim.x + threadIdx.x;
  if (i < n) d[i] = rsqrtf(fmaxf(d[i], 1.0f));   // clip(deg,1)^-0.5, in place
}

// ------- edge gather + scatter-add, float4 vectorized (nsrc folded into gather) -------
// One thread handles 4 consecutive features of one edge: b128 gather + 4 fp32 atomics.
__global__ void scatter4_kernel(const float4* __restrict__ x4, const int* __restrict__ src,
                                const int* __restrict__ dst, const float* __restrict__ nsrc,
                                float* __restrict__ agg, long long total, int d4, int d4sh) {
  long long t = (long long)blockIdx.x * blockDim.x + threadIdx.x;
  if (t >= total) return;
  int e  = (int)(t >> d4sh);
  int f4 = (int)(t & (long long)(d4 - 1));
  int s  = src[e];
  float ns = nsrc[s];
  float4 v = x4[(long long)s * d4 + f4];
  float* ag = agg + ((long long)dst[e] * d4 + f4) * 4;
  unsafeAtomicAdd(ag + 0, v.x * ns);
  unsafeAtomicAdd(ag + 1, v.y * ns);
  unsafeAtomicAdd(ag + 2, v.z * ns);
  unsafeAtomicAdd(ag + 3, v.w * ns);
}

// ---------------- fp32 WMMA GEMM: y = (agg * ndst) @ W + b ----------------
// 256-thread block = 8 waves. All waves share one 16-col N-tile whose W panel
// (d_in x 16, <= 8KB) is staged in LDS once; each wave computes one 16x16 tile
// via a K-loop of V_WMMA_F32_16X16X4_F32 fed from ds_load.
// A 16x4 layout: lane<16 -> M=lane, K={k,k+1}; lane>=16 -> M=lane-16, K={k+2,k+3}.
// B 4x16 mirrors the K striping. C/D: VGPR r -> M = r + 8*(lane>=16), N = lane&15.
#define MT_PER_BLOCK 8
__global__ void wmma_gemm_kernel(const float* __restrict__ agg, const float* __restrict__ ndst,
                                 const float* __restrict__ W, const float* __restrict__ bias,
                                 float* __restrict__ y, int d_in, int d_out, int mtiles) {
  __shared__ float Bs[128 * 16];                 // max d_in * 16 = 8KB

  const int tid  = threadIdx.x;
  const int colbase = blockIdx.y * 16;

  // cooperative fill of the W panel: Bs[k][n] = W[k*d_out + colbase + n]
  const int panel = d_in * 16;
  for (int i = tid; i < panel; i += 256) {
    int kk = i >> 4, nn = i & 15;
    Bs[i] = W[(long long)kk * d_out + colbase + nn];
  }
  __syncthreads();

  const int wave  = tid >> 5;                    // 0..7
  const int lane  = tid & 31;                    // wave32
  const int mtile = blockIdx.x * MT_PER_BLOCK + wave;
  if (mtile >= mtiles) return;                   // wave-uniform: EXEC stays full

  const int r16 = lane & 15;
  const int kh  = (lane >> 4) << 1;              // 0 or 2
  const long long arow = (long long)mtile * 16 + r16;
  const float nd = ndst[arow];
  const float* __restrict__ ap = agg + arow * (long long)d_in;

  v8f c = {0.f, 0.f, 0.f, 0.f, 0.f, 0.f, 0.f, 0.f};
  for (int k = 0; k < d_in; k += 4) {
    const int ka = k + kh;
    v2f a, b;
    a.x = ap[ka]     * nd;
    a.y = ap[ka + 1] * nd;
    b.x = Bs[(ka)     * 16 + r16];
    b.y = Bs[(ka + 1) * 16 + r16];
    c = __builtin_amdgcn_wmma_f32_16x16x4_f32(false, a, false, b, (short)0, c, false, false);
  }

  const int gcol = colbase + r16;
  const float bv = bias[gcol];
  const long long rbase = (long long)mtile * 16 + ((lane >> 4) << 3);
#pragma unroll
  for (int r = 0; r < 8; ++r)
    y[(rbase + r) * d_out + gcol] = c[r] + bv;
}

// ---------------- per-column sum / sumsq ----------------
__global__ void colstats_kernel(const float* __restrict__ y, float* __restrict__ stats,
                                int N, int d, int rows) {
  const int f = threadIdx.x;                      // blockDim.x == d
  long long r0 = (long long)blockIdx.x * rows;
  long long r1 = r0 + rows; if (r1 > N) r1 = N;
  float s = 0.f, s2 = 0.f;
  for (long long r = r0; r < r1; ++r) {
    float v = y[r * d + f];
    s += v; s2 += v * v;
  }
  unsafeAtomicAdd(&stats[f],     s);
  unsafeAtomicAdd(&stats[d + f], s2);
}

__global__ void bnparam_kernel(const float* __restrict__ stats, const float* __restrict__ g,
                               const float* __restrict__ beta, float* __restrict__ sb,
                               int N, int d) {
  int f = blockIdx.x * blockDim.x + threadIdx.x;
  if (f >= d) return;
  float invN = 1.0f / (float)N;
  float mu  = stats[f] * invN;
  float var = stats[d + f] * invN - mu * mu;       // biased variance
  float sc  = g[f] * rsqrtf(var + BN_EPS);
  sb[f]     = sc;
  sb[d + f] = beta[f] - mu * sc;
}

__global__ void bn_lrelu4_kernel(float4* __restrict__ y4, const float* __restrict__ sb,
                                 long long n4, int d4) {
  long long i = (long long)blockIdx.x * blockDim.x + threadIdx.x;
  if (i >= n4) return;
  int f = (int)((i & (long long)(d4 - 1)) << 2);
  int d = d4 << 2;
  float4 v = y4[i];
  float a0 = v.x * sb[f + 0] + sb[d + f + 0];
  float a1 = v.y * sb[f + 1] + sb[d + f + 1];
  float a2 = v.z * sb[f + 2] + sb[d + f + 2];
  float a3 = v.w * sb[f + 3] + sb[d + f + 3];
  v.x = a0 > 0.f ? a0 : SLOPE * a0;
  v.y = a1 > 0.f ? a1 : SLOPE * a1;
  v.z = a2 > 0.f ? a2 : SLOPE * a2;
  v.w = a3 > 0.f ? a3 : SLOPE * a3;
  y4[i] = v;
}

// ---------------- pooling ----------------
__global__ void poolcnt_kernel(const int* __restrict__ gid, float* __restrict__ cnt, int N) {
  int i = blockIdx.x * blockDim.x + threadIdx.x;
  if (i < N) unsafeAtomicAdd(&cnt[gid[i]], 1.0f);
}

// d == 32: one thread per (node, 4-feature chunk) -> b128 load + 4 atomics
__global__ void poolsum4_kernel(const float4* __restrict__ h4, const int* __restrict__ gid,
                                float* __restrict__ pooled, long long total) {
  long long t = (long long)blockIdx.x * blockDim.x + threadIdx.x;
  if (t >= total) return;
  int i  = (int)(t >> 3);
  int f4 = (int)(t & 7);
  float4 v = h4[(long long)i * 8 + f4];
  float* pg = pooled + ((long long)gid[i] * 8 + f4) * 4;
  unsafeAtomicAdd(pg + 0, v.x);
  unsafeAtomicAdd(pg + 1, v.y);
  unsafeAtomicAdd(pg + 2, v.z);
  unsafeAtomicAdd(pg + 3, v.w);
}

__global__ void final_kernel(const float* __restrict__ pooled, const float* __restrict__ cnt,
                             const float* __restrict__ Wc, const float* __restrict__ bc,
                             float* __restrict__ out, int G) {
  int t = blockIdx.x * blockDim.x + threadIdx.x;
  if (t >= G * 2) return;
  int g = t >> 1, c = t & 1;
  float inv = 1.0f / fmaxf(cnt[g], 1.0f);
  float acc = bc[c];
  for (int f = 0; f < 32; ++f)
    acc += pooled[g * 32 + f] * inv * Wc[f * 2 + c];
  out[t] = acc;
}

// ---------------- host side ----------------
extern "C" void kernel_launch(void* const* d_in, const int* in_sizes, int n_in,
                              void* d_out, int out_size, void* d_ws, size_t ws_size,
                              hipStream_t stream) {
  const float* feat  = (const float*)d_in[0];
  const int*   src   = (const int*)  d_in[1];
  const int*   dst   = (const int*)  d_in[2];
  const int*   gid   = (const int*)  d_in[3];
  const float* W1    = (const float*)d_in[4];
  const float* b1    = (const float*)d_in[5];
  const float* g1    = (const float*)d_in[6];
  const float* be1   = (const float*)d_in[7];
  const float* W2    = (const float*)d_in[8];
  const float* b2    = (const float*)d_in[9];
  const float* g2    = (const float*)d_in[10];
  const float* be2   = (const float*)d_in[11];
  const float* W3    = (const float*)d_in[12];
  const float* b3    = (const float*)d_in[13];
  const float* g3    = (const float*)d_in[14];
  const float* be3   = (const float*)d_in[15];
  const float* Wc    = (const float*)d_in[16];
  const float* bc    = (const float*)d_in[17];
  float*       out   = (float*)d_out;

  const int F_IN = 128;
  const int N = in_sizes[0] / F_IN;     // 50000 (divisible by 16)
  const int E = in_sizes[1];            // 800000
  const int G = out_size / 2;           // 64

  // workspace layout (floats)
  float* agg    = (float*)d_ws;                         // N*128
  float* yb     = agg    + (size_t)N * 128;             // N*128
  float* nsrcb  = yb     + (size_t)N * 128;             // N   (deg_out -> norm, in place)
  float* ndstb  = nsrcb  + (size_t)N;                   // N   (deg_in  -> norm, in place)
  float* stats  = ndstb  + (size_t)N;                   // 2*128
  float* sb     = stats  + 256;                         // 2*128
  float* pooled = sb     + 256;                         // G*32
  float* cnt    = pooled + (size_t)G * 32;              // G   (contiguous after pooled)

  auto zero = [&](float* p, long long n) {              // n must be multiple of 4 (all are)
    long long n4 = n >> 2;
    unsigned blocks = (unsigned)((n4 + 255) / 256);
    zero4_kernel<<<blocks, 256, 0, stream>>>((float4*)p, n4);
  };
  auto shift4_of = [](int d) { return d == 128 ? 5 : (d == 64 ? 4 : 3); };  // log2(d/4)

  // ---- degree norms (once) ----
  zero(nsrcb, 2LL * N);
  degree_kernel<<<(E + 255) / 256, 256, 0, stream>>>(src, dst, nsrcb, ndstb, E);
  norm_kernel<<<(2 * N + 255) / 256, 256, 0, stream>>>(nsrcb, 2LL * N);

  // ---- one GCN layer: agg = scatter(x*nsrc); y = (agg*ndst)@W + b; BN; LeakyReLU ----
  auto layer = [&](const float* x, int d_in, int d_out,
                   const float* W, const float* b, const float* g, const float* beta) {
    zero(agg, (long long)N * d_in);
    long long st = (long long)E * (d_in >> 2);
    scatter4_kernel<<<(unsigned)((st + 255) / 256), 256, 0, stream>>>(
        (const float4*)x, src, dst, nsrcb, agg, st, d_in >> 2, shift4_of(d_in));

    int mtiles = N / 16;
    dim3 ggrid((unsigned)((mtiles + MT_PER_BLOCK - 1) / MT_PER_BLOCK),
               (unsigned)(d_out / 16));
    wmma_gemm_kernel<<<ggrid, 256, 0, stream>>>(agg, ndstb, W, b, yb, d_in, d_out, mtiles);

    zero(stats, 2LL * d_out);
    const int ROWS = 128;
    colstats_kernel<<<(N + ROWS - 1) / ROWS, d_out, 0, stream>>>(yb, stats, N, d_out, ROWS);
    bnparam_kernel<<<1, d_out, 0, stream>>>(stats, g, beta, sb, N, d_out);

    long long n4 = ((long long)N * d_out) >> 2;
    bn_lrelu4_kernel<<<(unsigned)((n4 + 255) / 256), 256, 0, stream>>>(
        (float4*)yb, sb, n4, d_out >> 2);
  };

  layer(feat, 128, 128, W1, b1, g1, be1);   // h1 in yb [N,128]
  layer(yb,   128,  64, W2, b2, g2, be2);   // h2 in yb [N,64]
  layer(yb,    64,  32, W3, b3, g3, be3);   // h3 in yb [N,32]

  // ---- per-graph mean pooling + classifier ----
  zero(pooled, (long long)G * 32 + G);
  poolcnt_kernel<<<(N + 255) / 256, 256, 0, stream>>>(gid, cnt, N);
  long long pt = (long long)N * 8;          // N * (32/4) float4 chunks
  poolsum4_kernel<<<(unsigned)((pt + 255) / 256), 256, 0, stream>>>(
      (const float4*)yb, gid, pooled, pt);
  final_kernel<<<1, 128, 0, stream>>>(pooled, cnt, Wc, bc, out, G);
}